// RDESIRouter_32564442038661
// MI455X (gfx1250) — compile-verified
//
#include <hip/hip_runtime.h>
#include <hip/hip_bf16.h>
#include <math.h>

// Problem constants (from reference): B=4, S=4096 -> T=16384 tokens, H=4096, E=64, K=2
#define TOKENS   16384
#define HDIM     4096
#define NEXP     64
#define BK       64        // K-panel width staged in LDS
#define NPANELS  (HDIM / BK)
#define LDSPITCH 68        // floats per expert row: 272B = 16B-aligned, bank-conflict-free reads

typedef __attribute__((ext_vector_type(2))) float v2f;
typedef __attribute__((ext_vector_type(4))) float v4f;
typedef __attribute__((ext_vector_type(8))) float v8f;

// -------------------------------------------------------------------------
// Kernel 1: fused gate GEMM (fp32 WMMA 16x16x4) + additive-bias epilogue.
// Block = 256 threads = 8 waves; each wave owns a 16-row M-tile x all 64
// experts (4 N-tiles). gate_weight K-panels are staged in double-buffered
// LDS once per block and shared by all 8 waves (8x L2 traffic reduction);
// B fragments come from ds_load_b64, A fragments stream from HBM.
//
// A-fragment layout (ISA 7.12.2, 32-bit A 16x4):
//   lanes 0-15 : row = lane,    VGPR0/1 = cols k, k+1
//   lanes 16-31: row = lane-16, VGPR0/1 = cols k+2, k+3
// -------------------------------------------------------------------------
__global__ __launch_bounds__(256, 2) void rdesi_gate_gemm(
    const float* __restrict__ x, const float* __restrict__ W,
    const float* __restrict__ rep, const float* __restrict__ loads,
    const float* __restrict__ counts, const float* __restrict__ total,
    float* __restrict__ base_logits, float* __restrict__ sel_scores)
{
    __shared__ float wp[2][NEXP * LDSPITCH];   // 2 x 17.4 KB W panels

    const int tid   = (int)threadIdx.x;
    const int lane  = tid & 31;
    const int wave  = tid >> 5;
    const int lo    = lane & 15;
    const int hi    = lane >> 4;
    const int mBase = ((int)blockIdx.x * 8 + wave) * 16;

    const float* xrow = x + (size_t)(mBase + lo) * HDIM + (hi << 1);

    // Cooperative W-panel staging: thread -> (expert row, 16-col chunk).
    const int we = tid >> 2;             // expert row 0..63
    const int wc = (tid & 3) << 4;       // column offset 0,16,32,48
    const float* wsrc = W + (size_t)we * HDIM + wc;

    v4f wr[4];
    // Prologue: stage panel 0.
#pragma unroll
    for (int j = 0; j < 4; ++j) wr[j] = *(const v4f*)(wsrc + j * 4);
#pragma unroll
    for (int j = 0; j < 4; ++j)
        *(v4f*)(&wp[0][we * LDSPITCH + wc + j * 4]) = wr[j];
    __syncthreads();

    v8f acc0 = {}, acc1 = {}, acc2 = {}, acc3 = {};

    for (int p = 0; p < NPANELS; ++p) {
        const int kk = p * BK;
        // Fetch next panel into registers while computing on the current one.
        if (p + 1 < NPANELS) {
#pragma unroll
            for (int j = 0; j < 4; ++j)
                wr[j] = *(const v4f*)(wsrc + kk + BK + j * 4);
        }
        // Speculative prefetch of the A stream two panels ahead.
        __builtin_prefetch(xrow + kk + 2 * BK, 0, 1);

        const float* bp = &wp[p & 1][lo * LDSPITCH + (hi << 1)];
#pragma unroll
        for (int u = 0; u < BK; u += 4) {
            const v2f a  = *(const v2f*)(xrow + kk + u);
            const v2f b0 = *(const v2f*)(bp + u);
            const v2f b1 = *(const v2f*)(bp + 16 * LDSPITCH + u);
            const v2f b2 = *(const v2f*)(bp + 32 * LDSPITCH + u);
            const v2f b3 = *(const v2f*)(bp + 48 * LDSPITCH + u);
            acc0 = __builtin_amdgcn_wmma_f32_16x16x4_f32(false, a, false, b0, (short)0, acc0, false, false);
            acc1 = __builtin_amdgcn_wmma_f32_16x16x4_f32(false, a, false, b1, (short)0, acc1, false, false);
            acc2 = __builtin_amdgcn_wmma_f32_16x16x4_f32(false, a, false, b2, (short)0, acc2, false, false);
            acc3 = __builtin_amdgcn_wmma_f32_16x16x4_f32(false, a, false, b3, (short)0, acc3, false, false);
        }

        if (p + 1 < NPANELS) {
            // Store next panel into the other buffer; one barrier per panel
            // is sufficient (last readers of that buffer were fenced by the
            // previous barrier).
#pragma unroll
            for (int j = 0; j < 4; ++j)
                *(v4f*)(&wp[(p + 1) & 1][we * LDSPITCH + wc + j * 4]) = wr[j];
            __syncthreads();
        }
    }

    // Per-lane additive bias for expert e = nt*16 + (lane&15).
    // updated_loads == expert_loads exactly (alpha*l + (1-alpha)*l).
    const float lg = logf(total[0] + 1.0f);
    float bias[4];
#pragma unroll
    for (int nt = 0; nt < 4; ++nt) {
        const int e = nt * 16 + lo;
        bias[nt] = 0.1f * rep[e] - 0.1f * loads[e]
                 + 0.1f * sqrtf(lg / (counts[e] + 1e-10f));
    }

    // C/D layout: VGPR i holds (M = i + 8*hi, N = lo) for each N-tile.
#pragma unroll
    for (int i = 0; i < 8; ++i) {
        const size_t row = (size_t)(mBase + i + 8 * hi) * NEXP + (size_t)lo;
        const float v0 = acc0[i], v1 = acc1[i], v2 = acc2[i], v3 = acc3[i];
        base_logits[row +  0] = v0;
        base_logits[row + 16] = v1;
        base_logits[row + 32] = v2;
        base_logits[row + 48] = v3;
        sel_scores[row +  0] = v0 + bias[0];
        sel_scores[row + 16] = v1 + bias[1];
        sel_scores[row + 32] = v2 + bias[2];
        sel_scores[row + 48] = v3 + bias[3];
    }
}

// -------------------------------------------------------------------------
// Kernel 2: per-token top-2 + softmax over the 2 selected scores.
// Strict '>' scan preserves jax.lax.top_k's lowest-index tie-breaking.
// First 64 threads also emit updated_loads (== expert_loads).
// -------------------------------------------------------------------------
__global__ __launch_bounds__(256) void rdesi_topk(
    const float* __restrict__ sel, const float* __restrict__ loads,
    float* __restrict__ rw, float* __restrict__ eidx, float* __restrict__ upd)
{
    const int t = (int)(blockIdx.x * blockDim.x + threadIdx.x);
    if (t < NEXP) upd[t] = loads[t];
    if (t >= TOKENS) return;

    const float* row = sel + (size_t)t * NEXP;
    float best = -INFINITY, second = -INFINITY;
    int bi = 0, si = 0;
#pragma unroll 8
    for (int e = 0; e < NEXP; ++e) {
        const float v = row[e];
        if (v > best)        { second = best; si = bi; best = v; bi = e; }
        else if (v > second) { second = v; si = e; }
    }
    const float e1  = expf(second - best);
    const float inv = 1.0f / (1.0f + e1);
    rw[(size_t)t * 2 + 0]   = inv;
    rw[(size_t)t * 2 + 1]   = e1 * inv;
    eidx[(size_t)t * 2 + 0] = (float)bi;
    eidx[(size_t)t * 2 + 1] = (float)si;
}

extern "C" void kernel_launch(void* const* d_in, const int* in_sizes, int n_in,
                              void* d_out, int out_size, void* d_ws, size_t ws_size,
                              hipStream_t stream) {
    const float* x      = (const float*)d_in[0];  // [T, H]
    const float* W      = (const float*)d_in[1];  // [E, H]
    const float* rep    = (const float*)d_in[2];  // [E]
    const float* loads  = (const float*)d_in[3];  // [E]
    const float* counts = (const float*)d_in[4];  // [E]
    const float* total  = (const float*)d_in[5];  // [1]

    float* out  = (float*)d_out;
    float* rw   = out;                                    // [T,2]  = 32768
    float* eidx = out + (size_t)TOKENS * 2;               // [T,2]  = 32768
    float* blog = out + (size_t)TOKENS * 4;               // [T,E]  = 1048576
    float* sel  = blog + (size_t)TOKENS * NEXP;           // [T,E]  = 1048576
    float* upd  = sel  + (size_t)TOKENS * NEXP;           // [E]    = 64

    rdesi_gate_gemm<<<TOKENS / 128, 256, 0, stream>>>(x, W, rep, loads, counts, total, blog, sel);
    rdesi_topk<<<TOKENS / 256, 256, 0, stream>>>(sel, loads, rw, eidx, upd);
}